// LSTM_32100585570916
// MI455X (gfx1250) — compile-verified
//
#include <hip/hip_runtime.h>
#include <math.h>

#define VSIZE 32000
#define EMB   32
#define HID   32
#define BATCH 16
#define SEQ   256
#define LN_EPS 1e-5f

typedef __attribute__((ext_vector_type(2))) float v2f;
typedef __attribute__((ext_vector_type(8))) float v8f;

__device__ __forceinline__ float sigmoidf_(float x) {
    return 1.0f / (1.0f + __expf(-x));
}

// ---------------------------------------------------------------------------
// Kernel 1: embedding gather + LayerNorm-LSTM scan + output projection.
// One workgroup per batch element (16 independent recurrences), 128 threads =
// one thread per gate element (4 gates x 32 hidden). h/c/x live in LDS,
// w_ih row in registers. LayerNorm stats via wave32 shfl_xor + LDS combine.
// Writes proj = h_t @ w_proj^T  (shape (B*SEQ, EMB)) into workspace.
// ---------------------------------------------------------------------------
__global__ __launch_bounds__(128) void lstm_scan_kernel(
    const int*   __restrict__ tokens,   // (B, SEQ)
    const float* __restrict__ embed,    // (VSIZE, EMB)
    const float* __restrict__ w_ih,     // (4*HID, EMB+HID)
    const float* __restrict__ b_ih,     // (4*HID)
    const float* __restrict__ ln_w,     // (4, HID) flat 128
    const float* __restrict__ ln_b,     // (4, HID) flat 128
    const float* __restrict__ w_proj,   // (EMB, HID)
    const float* __restrict__ hidden0,  // (B, HID)
    const float* __restrict__ cell0,    // (B, HID)
    float*       __restrict__ proj_out) // (B*SEQ, EMB) workspace
{
    const int b    = blockIdx.x;
    const int t    = threadIdx.x;   // 0..127 : gate element index
    const int lane = t & 31;
    const int wave = t >> 5;

    __shared__ float x_lds[EMB];
    __shared__ float h_lds[HID];
    __shared__ float c_lds[HID];
    __shared__ float g_lds[128];
    __shared__ float red_s[4];
    __shared__ float red_s2[4];

    // per-thread weight row (64 floats in VGPRs)
    float w[EMB + HID];
    #pragma unroll
    for (int k = 0; k < EMB + HID; ++k) w[k] = w_ih[t * (EMB + HID) + k];

    // threads 0..31 also own one row of w_proj for the per-step projection
    float wp[HID];
    if (t < EMB) {
        #pragma unroll
        for (int k = 0; k < HID; ++k) wp[k] = w_proj[t * HID + k];
    }
    if (t < HID) {
        h_lds[t] = hidden0[b * HID + t];
        c_lds[t] = cell0[b * HID + t];
    }
    const float bih = b_ih[t];
    const float lnw = ln_w[t];
    const float lnb = ln_b[t];
    __syncthreads();

    for (int s = 0; s < SEQ; ++s) {
        // gather x_t = embed[token]
        const int tok = tokens[b * SEQ + s];
        if (t < EMB) x_lds[t] = embed[tok * EMB + t];
        __syncthreads();

        // g[t] = [x_t ; h] . w_ih[t,:] + b_ih[t]
        float acc = bih;
        #pragma unroll
        for (int k = 0; k < EMB; ++k) acc = fmaf(w[k],       x_lds[k], acc);
        #pragma unroll
        for (int k = 0; k < HID; ++k) acc = fmaf(w[EMB + k], h_lds[k], acc);

        // LayerNorm statistics over all 128 gate values of this sample
        float s1 = acc, s2 = acc * acc;
        #pragma unroll
        for (int off = 16; off > 0; off >>= 1) {
            s1 += __shfl_xor(s1, off, 32);
            s2 += __shfl_xor(s2, off, 32);
        }
        if (lane == 0) { red_s[wave] = s1; red_s2[wave] = s2; }
        __syncthreads();
        const float tot  = red_s[0]  + red_s[1]  + red_s[2]  + red_s[3];
        const float tot2 = red_s2[0] + red_s2[1] + red_s2[2] + red_s2[3];
        const float mu   = tot * (1.0f / 128.0f);
        const float var  = tot2 * (1.0f / 128.0f) - mu * mu;
        const float inv  = rsqrtf(var + LN_EPS);
        g_lds[t] = (acc - mu) * inv * lnw + lnb;
        __syncthreads();

        // gate nonlinearities + state update (threads 0..31, one per hid idx)
        if (t < HID) {
            const float gi = g_lds[t];
            const float gf = g_lds[32 + t];
            const float gh = g_lds[64 + t];
            const float go = g_lds[96 + t];
            float cc = sigmoidf_(gf) * c_lds[t] + sigmoidf_(gi) * tanhf(gh);
            c_lds[t] = cc;
            h_lds[t] = sigmoidf_(go) * cc;
        }
        __syncthreads();

        // proj[b,s,e] = h . w_proj[e,:]
        if (t < EMB) {
            float p = 0.0f;
            #pragma unroll
            for (int k = 0; k < HID; ++k) p = fmaf(wp[k], h_lds[k], p);
            proj_out[(b * SEQ + s) * EMB + t] = p;
        }
        // no trailing barrier needed: next-iter writers (x_lds) only race with
        // readers that already passed the stats barrier this iteration.
    }
}

// ---------------------------------------------------------------------------
// Kernel 2: logits = proj @ embed^T + b_out, fp32 WMMA.
// M=4096, N=32000, K=32. Block = 256 threads = 8 waves; block computes a
// 16(M) x 128(N) strip: each wave owns one 16x16 tile, K=32 as 8 chained
// V_WMMA_F32_16X16X4_F32. A tile staged in LDS (ds_load_b64 fragments),
// B fragments loaded straight from embed (4MB -> L2 resident).
// Output is write-once 524MB -> nontemporal stores.
//
// Fragment layouts (ISA 7.12.2):
//   A 16x4 f32 : lane L, vgpr j -> A[L%16][(L/16)*2 + j]
//   B 4x16 f32 : lane L, vgpr j -> B[(L/16)*2 + j][L%16]   (B[k][n]=embed[n][k])
//   D 16x16    : lane L, vgpr j -> D[j + (L/16)*8][L%16]
// ---------------------------------------------------------------------------
__global__ __launch_bounds__(256) void logits_gemm_kernel(
    const float* __restrict__ proj,   // (4096, 32)
    const float* __restrict__ embed,  // (VSIZE, 32)
    const float* __restrict__ b_out,  // (VSIZE)
    float*       __restrict__ out)    // (4096, VSIZE)
{
    __shared__ float a_lds[16 * EMB];   // 16-row A tile (contiguous slice of proj)

    const int t    = threadIdx.x;
    const int lane = t & 31;
    const int wave = t >> 5;
    const int m0   = blockIdx.y * 16;
    const int n0   = blockIdx.x * 128 + wave * 16;

    // stage A tile: rows m0..m0+15 of proj are one contiguous 512-float slice
    a_lds[t]       = proj[m0 * EMB + t];
    a_lds[t + 256] = proj[m0 * EMB + t + 256];
    __syncthreads();

    const int mrow  = lane & 15;         // A row within tile / D column index n
    const int khalf = (lane >> 4) * 2;   // K sub-offset selected by lane half
    const int nn    = n0 + (lane & 15);  // output column for this lane

    v8f acc = {};
    #pragma unroll
    for (int kc = 0; kc < 8; ++kc) {
        const int kb = kc * 4 + khalf;
        v2f afrag = *(const v2f*)&a_lds[mrow * EMB + kb];      // ds_load_b64
        v2f bfrag = *(const v2f*)&embed[nn * EMB + kb];        // global_load_b64
        acc = __builtin_amdgcn_wmma_f32_16x16x4_f32(
            /*neg_a=*/false, afrag, /*neg_b=*/false, bfrag,
            /*c_mod=*/(short)0, acc, /*reuse_a=*/false, /*reuse_b=*/false);
    }

    const float bias = b_out[nn];
    #pragma unroll
    for (int j = 0; j < 8; ++j) {
        const int m = m0 + j + (lane >> 4) * 8;
        __builtin_nontemporal_store(acc[j] + bias,
                                    &out[(size_t)m * VSIZE + nn]);
    }
}

// ---------------------------------------------------------------------------
extern "C" void kernel_launch(void* const* d_in, const int* in_sizes, int n_in,
                              void* d_out, int out_size, void* d_ws, size_t ws_size,
                              hipStream_t stream) {
    const int*   tokens  = (const int*)  d_in[0];
    const float* embed   = (const float*)d_in[1];
    const float* w_ih    = (const float*)d_in[2];
    const float* b_ih    = (const float*)d_in[3];
    const float* ln_w    = (const float*)d_in[4];
    const float* ln_b    = (const float*)d_in[5];
    const float* w_proj  = (const float*)d_in[6];
    const float* b_out   = (const float*)d_in[7];
    const float* hidden0 = (const float*)d_in[8];
    const float* cell0   = (const float*)d_in[9];

    float* proj_ws = (float*)d_ws;             // (B*SEQ, EMB) = 512 KB
    float* out     = (float*)d_out;            // (B*SEQ, VSIZE)

    // Phase 1: sequential LSTM scan, one workgroup per batch row.
    lstm_scan_kernel<<<dim3(BATCH), dim3(128), 0, stream>>>(
        tokens, embed, w_ih, b_ih, ln_w, ln_b, w_proj, hidden0, cell0, proj_ws);

    // Phase 2: big WMMA GEMM. N: 32000/128 = 250 blocks, M: 4096/16 = 256.
    logits_gemm_kernel<<<dim3(VSIZE / 128, (BATCH * SEQ) / 16), dim3(256), 0, stream>>>(
        proj_ws, embed, b_out, out);
}